// SFAM_43490838839386
// MI455X (gfx1250) — compile-verified
//
#include <hip/hip_runtime.h>

typedef __attribute__((ext_vector_type(2))) float v2f;
typedef __attribute__((ext_vector_type(8))) float v8f;

#define D_DIM 256
#define S_DIM 256
#define NOISE_STD 0.01f
#define EPS_NORM 1e-12f

// One block per sample b. 256 threads = 8 wave32s.
// Phase 1: f = clamp(tanh(raw)/||tanh(raw)||_2 + 0.01*noise, -1.5, 1.5) -> LDS
// Phase 2: hashed = f * P_b via V_WMMA_F32_16X16X4_F32 accumulation,
//          out = tanh(hashed). P_b (256KB) streamed once, non-temporal.
__global__ __launch_bounds__(256)
void sfam_fused_wmma(const float* __restrict__ raw,
                     const float* __restrict__ proj,
                     const float* __restrict__ noise,
                     float* __restrict__ out) {
  const int b    = blockIdx.x;
  const int tid  = threadIdx.x;
  const int lane = tid & 31;
  const int w    = tid >> 5;            // wave id 0..7

  __shared__ float sF[D_DIM];
  __shared__ float sRed[8];

  // ---------- Phase 1: tanh + L2 norm + noise + clamp ----------
  const size_t rowBase = (size_t)b * D_DIM;
  const float t = tanhf(raw[rowBase + tid]);

  // wave32 butterfly reduction of t^2, then one cross-wave LDS step
  float s2 = t * t;
  #pragma unroll
  for (int m = 16; m > 0; m >>= 1) s2 += __shfl_xor(s2, m, 32);
  if (lane == 0) sRed[w] = s2;
  __syncthreads();
  float tot = 0.f;
  #pragma unroll
  for (int i = 0; i < 8; ++i) tot += sRed[i];

  const float nrm = fmaxf(sqrtf(tot), EPS_NORM);
  float f = t / nrm + NOISE_STD * noise[rowBase + tid];
  f = fminf(fmaxf(f, -1.5f), 1.5f);
  __syncthreads();              // sRed reads done before sF overwrites nothing,
  sF[tid] = f;                  // but keep write ordered vs. phase-2 reads
  __syncthreads();

  // ---------- Phase 2: GEMV via WMMA f32 16x16x4 ----------
  // Wave w owns N-tiles 2w and 2w+1 (columns 32w .. 32w+31).
  // A (16x4 f32, 2 VGPRs): only row M=0 is live:
  //   lane 0  -> {f[k],   f[k+1]}   lane 16 -> {f[k+2], f[k+3]}   others 0.
  // Branch-free build: uniform ds_load_b64 of the f-chunk for ALL lanes
  // (lanes 0-15 read float2 at k/2, lanes 16-31 at k/2+1), then scale by a
  // per-lane selector (1.0 for lanes 0/16, else 0.0). No EXEC divergence.
  // B (4x16 f32, 2 VGPRs): lanes 0-15 = rows k,k+1; lanes 16-31 = rows k+2,k+3.
  const float* __restrict__ Pb = proj + (size_t)b * (D_DIM * (size_t)S_DIM);
  const v2f*   __restrict__ sF2 = (const v2f*)sF;

  const int   hi     = lane >> 4;                       // 0 or 1
  const int   colA   = (w * 2) * 16 + (lane & 15);
  const int   colB   = colA + 16;
  const int   rowOff = hi * 2;                          // 0 or 2
  const float sel    = ((lane & 15) == 0) ? 1.0f : 0.0f;

  v8f c0 = {0.f, 0.f, 0.f, 0.f, 0.f, 0.f, 0.f, 0.f};
  v8f c1 = {0.f, 0.f, 0.f, 0.f, 0.f, 0.f, 0.f, 0.f};

  #pragma unroll 8
  for (int k = 0; k < D_DIM; k += 4) {
    v2f fv = sF2[(k >> 1) + hi];        // uniform LDS read, broadcast
    v2f A;
    A.x = fv.x * sel;
    A.y = fv.y * sel;

    const float* __restrict__ pr = Pb + (size_t)(k + rowOff) * S_DIM;
    v2f B0, B1;
    B0.x = __builtin_nontemporal_load(pr + colA);          // row k+rowOff
    B0.y = __builtin_nontemporal_load(pr + S_DIM + colA);  // row k+rowOff+1
    B1.x = __builtin_nontemporal_load(pr + colB);
    B1.y = __builtin_nontemporal_load(pr + S_DIM + colB);

    c0 = __builtin_amdgcn_wmma_f32_16x16x4_f32(
        /*neg_a=*/false, A, /*neg_b=*/false, B0,
        /*c_mod=*/(short)0, c0, /*reuse_a=*/false, /*reuse_b=*/false);
    c1 = __builtin_amdgcn_wmma_f32_16x16x4_f32(
        /*neg_a=*/false, A, /*neg_b=*/false, B1,
        /*c_mod=*/(short)0, c1, /*reuse_a=*/false, /*reuse_b=*/false);
  }

  // C/D layout: VGPR0, lanes 0-15 hold M=0, N=lane -> row 0 of the result tile.
  if (lane < 16) {
    out[rowBase + colA] = tanhf(c0[0]);
    out[rowBase + colB] = tanhf(c1[0]);
  }
}

extern "C" void kernel_launch(void* const* d_in, const int* in_sizes, int n_in,
                              void* d_out, int out_size, void* d_ws, size_t ws_size,
                              hipStream_t stream) {
  const float* raw   = (const float*)d_in[0];  // (B, 256) f32
  const float* proj  = (const float*)d_in[1];  // (B, 256, 256) f32
  const float* noise = (const float*)d_in[2];  // (B, 256) f32
  float* out = (float*)d_out;                  // (B, 256) f32

  const int B = in_sizes[0] / D_DIM;           // 4096
  sfam_fused_wmma<<<B, 256, 0, stream>>>(raw, proj, noise, out);
}